// SoftDTWAlignmentCuda_69758858822518
// MI455X (gfx1250) — compile-verified
//
#include <hip/hip_runtime.h>
#include <hip/hip_bf16.h>

#define BB 8
#define MM 256
#define NN 256
#define DD 512
#define INF_ 1e8f

typedef float v2f __attribute__((ext_vector_type(2)));
typedef float v8f __attribute__((ext_vector_type(8)));

// ---------------------------------------------------------------------------
// 1) Inverse row norms: one wave32 per row of 512 floats.
// ---------------------------------------------------------------------------
__global__ void rownorm_kernel(const float* __restrict__ src,
                               float* __restrict__ invn, int nrows) {
    int row  = blockIdx.x * (blockDim.x >> 5) + (threadIdx.x >> 5);
    int lane = threadIdx.x & 31;
    if (row >= nrows) return;
    const float* p = src + (size_t)row * DD;
    float s = 0.f;
#pragma unroll
    for (int k = lane; k < DD; k += 32) { float v = p[k]; s += v * v; }
#pragma unroll
    for (int off = 16; off > 0; off >>= 1) s += __shfl_xor(s, off, 32);
    if (lane == 0) invn[row] = 1.0f / fmaxf(sqrtf(s), 1e-8f);
}

// ---------------------------------------------------------------------------
// 2) cost[b,m,n] = 1 - (x[b,m,:].y[b,n,:]) * invnx * invny  via fp32 WMMA.
//    One wave per 16x16 tile; K=512 in steps of 4 (v_wmma_f32_16x16x4_f32).
//    A 16x4 layout: lane%16 = M row, lane/16 selects K pair {2h, 2h+1}.
//    B 4x16 layout: lane%16 = N col, same K-pair split -> contiguous b64 loads.
// ---------------------------------------------------------------------------
__global__ void cost_wmma_kernel(const float* __restrict__ x,
                                 const float* __restrict__ y,
                                 const float* __restrict__ invnx,
                                 const float* __restrict__ invny,
                                 float* __restrict__ cost) {
    int wave = threadIdx.x >> 5;
    int lane = threadIdx.x & 31;
    int tile = blockIdx.x * (blockDim.x >> 5) + wave;   // 0..255 per batch
    int b    = blockIdx.y;
    int tm   = (tile >> 4) << 4;
    int tn   = (tile & 15) << 4;
    int hf   = lane >> 4;        // 0/1: which K pair
    int l16  = lane & 15;

    const float* xp = x + ((size_t)b * MM + tm + l16) * DD + (hf << 1);
    const float* yp = y + ((size_t)b * NN + tn + l16) * DD + (hf << 1);

    v8f acc = {};
#pragma unroll 4
    for (int k = 0; k < DD; k += 4) {
        v2f av = *(const v2f*)(xp + k);
        v2f bv = *(const v2f*)(yp + k);
        acc = __builtin_amdgcn_wmma_f32_16x16x4_f32(
            /*neg_a=*/false, av, /*neg_b=*/false, bv,
            /*c_mod=*/(short)0, acc, /*reuse_a=*/false, /*reuse_b=*/false);
    }

    // C/D layout: VGPR r holds (M = tm + r + 8*hf, N = tn + l16)
    float sy = invny[b * NN + tn + l16];
#pragma unroll
    for (int r = 0; r < 8; ++r) {
        int m = tm + r + (hf << 3);
        float sx = invnx[b * MM + m];
        cost[((size_t)b * MM + m) * NN + tn + l16] = 1.0f - acc[r] * sx * sy;
    }
}

// ---------------------------------------------------------------------------
// 3) Soft-DTW forward: anti-diagonal wavefront, thread i = row i.
//    LDS s1/s2 hold the previous two diagonals of R; own-left kept in reg.
//    Writes interior of R (rows/cols 1..M,1..N) to ws; distance to dist[b].
// ---------------------------------------------------------------------------
__global__ void dtw_fwd_kernel(const float* __restrict__ cost,
                               const float* __restrict__ gamma,
                               float* __restrict__ R,
                               float* __restrict__ dist) {
    const int M = MM, N = NN;
    int b = blockIdx.x;
    int i = threadIdx.x;
    float gv    = fmaxf(fabsf(gamma[0]), 1e-4f);
    float inv_g = 1.0f / gv;
    __shared__ float s1[MM];   // diagonal d-1 values, per row-thread
    __shared__ float s2[MM];   // diagonal d-2 values
    const float* cb = cost + (size_t)b * M * N;
    float* Rb = R + (size_t)b * (M + 1) * (N + 1);
    float prev = 0.f;          // this thread's value from previous diagonal
    for (int d = 0; d < M + N - 1; ++d) {
        int j = d - i;
        bool active = (j >= 0) & (j < N);
        float nv = 0.f;
        if (active) {
            float a, bb, c;
            if (i == 0) { a = (j == 0) ? 0.f : INF_; bb = INF_; }
            else        { a = (j == 0) ? INF_ : s2[i - 1]; bb = s1[i - 1]; }
            c = (j == 0) ? INF_ : prev;
            float mn  = fminf(a, fminf(bb, c));
            float sum = __expf((mn - a)  * inv_g)
                      + __expf((mn - bb) * inv_g)
                      + __expf((mn - c)  * inv_g);
            nv = cb[i * N + j] + (mn - gv * __logf(sum));
            Rb[(i + 1) * (N + 1) + (j + 1)] = nv;
        }
        __syncthreads();                       // all reads of s1/s2 done
        if (active) { s2[i] = prev; s1[i] = nv; prev = nv; }
        __syncthreads();                       // writes visible next step
    }
    if (i == M - 1) dist[b] = prev;            // R[M][N]
}

// ---------------------------------------------------------------------------
// 4) Soft-DTW backward: reverse wavefront. E borders are analytically 0
//    (only E[M][N]=1 preset), so alignment is written straight to d_out.
// ---------------------------------------------------------------------------
__global__ void dtw_bwd_kernel(const float* __restrict__ cost,
                               const float* __restrict__ gamma,
                               const float* __restrict__ R,
                               float* __restrict__ out) {
    const int M = MM, N = NN;
    int b = blockIdx.x;
    int i = threadIdx.x;
    float gv    = fmaxf(fabsf(gamma[0]), 1e-4f);
    float inv_g = 1.0f / gv;
    __shared__ float e1[MM];   // diagonal d+1 E values
    __shared__ float e2[MM];   // diagonal d+2 E values
    const float* cb = cost + (size_t)b * M * N;
    const float* Rb = R + (size_t)b * (M + 1) * (N + 1);
    float* ob = out + (size_t)b * M * N;

    for (int t = i; t < M * N; t += blockDim.x) ob[t] = 0.f;   // d_out is poisoned
    __syncthreads();
    if (i == M - 1) ob[(M - 1) * N + (N - 1)] = 1.f;           // E[M][N] = 1
    e1[i] = (i == M - 1) ? 1.f : 0.f;
    e2[i] = 0.f;
    float prev = (i == M - 1) ? 1.f : 0.f;
    __syncthreads();

    for (int d = M + N - 3; d >= 0; --d) {
        int j = d - i;
        bool active = (j >= 0) & (j < N);
        float nv = 0.f;
        if (active) {
            int ip = i + 1, jp = j + 1;
            float rc  = Rb[ip * (N + 1) + jp];
            float acc = 0.f;
            if ((i + 1 < M) & (j + 1 < N)) {   // diagonal successor
                float t = (Rb[(ip + 1) * (N + 1) + jp + 1]
                           - cb[(i + 1) * N + (j + 1)] - rc) * inv_g;
                acc += e2[i + 1] * __expf(fminf(fmaxf(t, -50.f), 50.f));
            }
            if (i + 1 < M) {                   // down successor
                float t = (Rb[(ip + 1) * (N + 1) + jp]
                           - cb[(i + 1) * N + j] - rc) * inv_g;
                acc += e1[i + 1] * __expf(fminf(fmaxf(t, -50.f), 50.f));
            }
            if (j + 1 < N) {                   // right successor (own prev)
                float t = (Rb[ip * (N + 1) + jp + 1]
                           - cb[i * N + (j + 1)] - rc) * inv_g;
                acc += prev * __expf(fminf(fmaxf(t, -50.f), 50.f));
            }
            nv = acc;
            ob[i * N + j] = nv;
        }
        __syncthreads();
        if (active) { e2[i] = prev; e1[i] = nv; prev = nv; }
        __syncthreads();
    }
}

// ---------------------------------------------------------------------------
extern "C" void kernel_launch(void* const* d_in, const int* in_sizes, int n_in,
                              void* d_out, int out_size, void* d_ws, size_t ws_size,
                              hipStream_t stream) {
    const float* x     = (const float*)d_in[0];   // (8,256,512) f32
    const float* y     = (const float*)d_in[1];   // (8,256,512) f32
    const float* gamma = (const float*)d_in[2];   // scalar f32
    float* out = (float*)d_out;                   // alignment (8*256*256) ++ distance (8)

    // workspace layout (floats): invnx | invny | cost | R
    float* invnx = (float*)d_ws;                              // 2048
    float* invny = invnx + (size_t)BB * MM;                   // 2048
    float* cost  = invny + (size_t)BB * NN;                   // 8*256*256
    float* R     = cost  + (size_t)BB * MM * NN;              // 8*257*257

    rownorm_kernel<<<dim3((BB * MM) / 8), 256, 0, stream>>>(x, invnx, BB * MM);
    rownorm_kernel<<<dim3((BB * NN) / 8), 256, 0, stream>>>(y, invny, BB * NN);

    // 256 tiles per batch, 4 waves (128 threads) per block -> 64 blocks x 8 batches
    cost_wmma_kernel<<<dim3(64, BB), 128, 0, stream>>>(x, y, invnx, invny, cost);

    dtw_fwd_kernel<<<dim3(BB), MM, 0, stream>>>(cost, gamma, R,
                                                out + (size_t)BB * MM * NN);
    dtw_bwd_kernel<<<dim3(BB), MM, 0, stream>>>(cost, gamma, R, out);
}